// TinyModel_3100966387817
// MI455X (gfx1250) — compile-verified
//
#include <hip/hip_runtime.h>
#include <hip/hip_bf16.h>

// ---------------------------------------------------------------------------
// Greedy NMS, IoU threshold 0.2, N = 8192 boxes.
//   1) nms_mask_kernel : full-GPU pairwise IoU -> bit mask (8192 x 256 words),
//                        column boxes staged in LDS via CDNA5 async-to-LDS.
//   2) nms_scan_kernel : 1 workgroup sequential greedy suppression over the
//                        mask; 4-deep row ring buffer in LDS fed by ASYNCcnt-
//                        tracked async loads; final scatter to original order.
//   3) nms_sort_kernel : 1 workgroup LDS bitonic argsort of scores (desc).
// ---------------------------------------------------------------------------

#define NMS_N   8192
#define NMS_NW  (NMS_N / 32)        // 256 mask words per row
#define IOU_THR 0.2f

// ---- CDNA5 async global->LDS support (guarded; falls back cleanly) --------
#if defined(__gfx1250__) && __has_builtin(__builtin_amdgcn_global_load_async_to_lds_b32)
#define HAS_ASYNC 1
#else
#define HAS_ASYNC 0
#endif

#if __has_builtin(__builtin_amdgcn_s_wait_asynccnt)
#define WAIT_ASYNC(n) __builtin_amdgcn_s_wait_asynccnt(n)
#else
#define WAIT_ASYNC(n) asm volatile("s_wait_asynccnt %0" ::"n"(n) : "memory")
#endif

#if HAS_ASYNC
typedef __attribute__((address_space(1))) int as1_int;
typedef __attribute__((address_space(3))) int as3_int;
__device__ __forceinline__ void async_ld_b32(const void* gsrc, void* ldst) {
  // global -> LDS async copy, tracked by ASYNCcnt
  __builtin_amdgcn_global_load_async_to_lds_b32((as1_int*)gsrc, (as3_int*)ldst,
                                                /*offset=*/0, /*cpol=*/0);
}
#endif

// ---------------------------------------------------------------------------
// Kernel: pairwise IoU bitmask.  gridDim = (256 col-blocks, 32 row-groups),
// 256 threads: thread = one sorted row, produces one 32-bit mask word.
// ---------------------------------------------------------------------------
__global__ __launch_bounds__(256) void nms_mask_kernel(
    const float4* __restrict__ boxes, unsigned* __restrict__ mask) {
  __shared__ float4 cbox[32];  // 512 B of column boxes
  const int tid = threadIdx.x;
  const int c0  = blockIdx.x * 32;
  const int row = blockIdx.y * 256 + tid;

#if HAS_ASYNC
  if (tid < 128) {  // 128 dwords = 32 float4 boxes
    float*       dst = ((float*)cbox) + tid;
    const float* src = ((const float*)(boxes + c0)) + tid;
    async_ld_b32((const void*)src, (void*)dst);
  }
  WAIT_ASYNC(0);
  __syncthreads();
#else
  if (tid < 32) cbox[tid] = boxes[c0 + tid];
  __syncthreads();
#endif

  const float4 rb = boxes[row];
  const float rArea = (rb.z - rb.x) * (rb.w - rb.y);

  unsigned word = 0u;
#pragma unroll
  for (int jj = 0; jj < 32; ++jj) {
    const int j = c0 + jj;
    const float4 cb = cbox[jj];
    float iw = fminf(rb.z, cb.z) - fmaxf(rb.x, cb.x);
    float ih = fminf(rb.w, cb.w) - fmaxf(rb.y, cb.y);
    iw = fmaxf(iw, 0.0f);
    ih = fmaxf(ih, 0.0f);
    const float inter = iw * ih;
    const float cArea = (cb.z - cb.x) * (cb.w - cb.y);
    const float uni   = rArea + cArea - inter;
    const bool  over  = (inter / fmaxf(uni, 1e-9f)) > IOU_THR;
    if (over && (j > row)) word |= (1u << jj);
  }
  mask[row * NMS_NW + blockIdx.x] = word;
}

// ---------------------------------------------------------------------------
// Kernel: sequential greedy scan (1 workgroup, 256 threads = 1 word each).
// Mask rows flow through a 4-deep LDS ring buffer filled by async loads;
// ASYNCcnt<=3 retires the oldest in-flight row (async loads complete in
// order per wave).  Then scatter surviving scores to original order.
// ---------------------------------------------------------------------------
__global__ __launch_bounds__(256) void nms_scan_kernel(
    const unsigned* __restrict__ mask, const int* __restrict__ s_idx,
    const float* __restrict__ scores, float* __restrict__ out) {
  __shared__ unsigned removed[NMS_NW];
  const int tid = threadIdx.x;
  removed[tid] = 0u;

#if HAS_ASYNC
  __shared__ unsigned rowbuf[4][NMS_NW];
  // prologue: launch rows 0..3 into the ring
  async_ld_b32((const void*)&mask[0 * NMS_NW + tid], (void*)&rowbuf[0][tid]);
  async_ld_b32((const void*)&mask[1 * NMS_NW + tid], (void*)&rowbuf[1][tid]);
  async_ld_b32((const void*)&mask[2 * NMS_NW + tid], (void*)&rowbuf[2][tid]);
  async_ld_b32((const void*)&mask[3 * NMS_NW + tid], (void*)&rowbuf[3][tid]);
#endif
  __syncthreads();

  for (int i = 0; i < NMS_N; ++i) {
#if HAS_ASYNC
    const int buf = i & 3;
    // retire the oldest in-flight row (row i); keep up to 3 rows in flight
    if (i < NMS_N - 3)            WAIT_ASYNC(3);
    else if (i == NMS_N - 3)      WAIT_ASYNC(2);
    else if (i == NMS_N - 2)      WAIT_ASYNC(1);
    else                          WAIT_ASYNC(0);

    const unsigned kw = removed[i >> 5];    // keep bit of row i is final by now
    const unsigned rv = rowbuf[buf][tid];   // consume slot before refilling it
    __syncthreads();
    if (!((kw >> (i & 31)) & 1u)) removed[tid] |= rv;
    if (i + 4 < NMS_N)
      async_ld_b32((const void*)&mask[(i + 4) * NMS_NW + tid],
                   (void*)&rowbuf[buf][tid]);
    __syncthreads();
#else
    if (i + 1 < NMS_N) __builtin_prefetch(&mask[(i + 1) * NMS_NW + tid], 0, 0);
    const unsigned kw = removed[i >> 5];
    __syncthreads();
    if (!((kw >> (i & 31)) & 1u)) removed[tid] |= mask[i * NMS_NW + tid];
    __syncthreads();
#endif
  }

  // scatter: out[orig] = kept ? score[orig] : 0
  for (int i = tid; i < NMS_N; i += 256) {
    const bool keep = !((removed[i >> 5] >> (i & 31)) & 1u);
    const int orig = s_idx[i];
    out[orig] = keep ? scores[orig] : 0.0f;
  }
}

// ---------------------------------------------------------------------------
// Kernel: bitonic argsort (descending by score) fully in LDS (64 KB),
// then gathers boxes into sorted order.
// ---------------------------------------------------------------------------
__global__ __launch_bounds__(1024) void nms_sort_kernel(
    const float* __restrict__ rois, const float* __restrict__ scores,
    int* __restrict__ s_idx, float4* __restrict__ boxes) {
  __shared__ float key[NMS_N];
  __shared__ int   idx[NMS_N];
  const int tid = threadIdx.x;

  for (int i = tid; i < NMS_N; i += 1024) { key[i] = scores[i]; idx[i] = i; }
  __syncthreads();

  for (int k = 2; k <= NMS_N; k <<= 1) {
    for (int j = k >> 1; j > 0; j >>= 1) {
      for (int i = tid; i < NMS_N; i += 1024) {
        const int ixj = i ^ j;
        if (ixj > i) {
          const bool up = ((i & k) == 0);
          const float ki = key[i], kj = key[ixj];
          // overall DESCENDING sort
          const bool sw = up ? (ki < kj) : (ki > kj);
          if (sw) {
            key[i] = kj; key[ixj] = ki;
            const int t = idx[i]; idx[i] = idx[ixj]; idx[ixj] = t;
          }
        }
      }
      __syncthreads();
    }
  }

  const float4* r4 = (const float4*)rois;
  for (int i = tid; i < NMS_N; i += 1024) {
    const int si = idx[i];
    s_idx[i] = si;
    boxes[i] = r4[si];  // gather boxes into sorted order
  }
}

// ---------------------------------------------------------------------------
extern "C" void kernel_launch(void* const* d_in, const int* in_sizes, int n_in,
                              void* d_out, int out_size, void* d_ws, size_t ws_size,
                              hipStream_t stream) {
  const float* rois   = (const float*)d_in[0];  // (N,4) f32
  const float* scores = (const float*)d_in[1];  // (N,)  f32
  float* out = (float*)d_out;

  char* ws = (char*)d_ws;
  int*      s_idx = (int*)ws;                           //  32 KB
  float4*   boxes = (float4*)(ws + 32768);              // 128 KB
  unsigned* mask  = (unsigned*)(ws + 32768 + 131072);   //   8 MB

  nms_sort_kernel<<<1, 1024, 0, stream>>>(rois, scores, s_idx, boxes);
  nms_mask_kernel<<<dim3(NMS_NW, NMS_N / 256), 256, 0, stream>>>(boxes, mask);
  nms_scan_kernel<<<1, 256, 0, stream>>>(mask, s_idx, scores, out);
}